// BilateralBlur_34505767256656
// MI455X (gfx1250) — compile-verified
//
#include <hip/hip_runtime.h>
#include <cstdint>

#define IMG_H 512
#define IMG_W 512
#define CH    3
#define TILE  32
#define PAD   3
#define LDSD      (TILE + 2 * PAD)   // 38
#define LDSPLANE  (LDSD * LDSD)      // 1444
#define NLDS      (CH * LDSPLANE)    // 4332 floats = 17328 B

__device__ __forceinline__ int reflect512(int i) {
    // jnp.pad mode="reflect": padded[-1] = x[1], padded[H] = x[H-2]
    i = (i < 0) ? -i : i;
    i = (i > IMG_H - 1) ? (2 * (IMG_H - 1) - i) : i;
    return i;
}

__global__ __launch_bounds__(256)
void bilateral7x7_kernel(const float* __restrict__ in, float* __restrict__ out) {
    __shared__ float smem[NLDS];

    const int tid    = threadIdx.x;
    const int tileX0 = blockIdx.x * TILE;
    const int tileY0 = blockIdx.y * TILE;
    const int b      = blockIdx.z;

    // ---- Stage (TILE+6)^2 x 3ch tile into LDS via CDNA5 async global->LDS ----
    // Per-lane GLOBAL_LOAD_ASYNC_TO_LDS_B32 (GV mode): LDS byte address in the
    // "vdst" VGPR (low 32 bits of the flat pointer == AS(3) offset per ISA
    // 10.2 aperture rules), 64-bit global address in a VGPR pair.
    // Split per channel so the index math is a single constant div by 38.
    const uint32_t lds_base = (uint32_t)(uintptr_t)smem;
    unsigned long long gbase =
        (unsigned long long)(uintptr_t)(in + (size_t)b * CH * IMG_H * IMG_W);

    for (int c = 0; c < CH; ++c) {
        for (int i = tid; i < LDSPLANE; i += 256) {
            int ly = i / LDSD;              // const div: mulhi + shift
            int lx = i - ly * LDSD;
            int gy = reflect512(tileY0 - PAD + ly);
            int gx = reflect512(tileX0 - PAD + lx);
            unsigned long long gaddr = gbase + 4ull * ((unsigned)(gy * IMG_W + gx));
            uint32_t ldsaddr = lds_base + 4u * (uint32_t)(c * LDSPLANE + i);
            asm volatile("global_load_async_to_lds_b32 %0, %1, off"
                         :: "v"(ldsaddr), "v"(gaddr)
                         : "memory");
        }
        gbase += 4ull * IMG_H * IMG_W;
    }
    // Each wave drains its own ASYNCcnt; the workgroup barrier then makes the
    // whole tile visible to every wave.
    asm volatile("s_wait_asynccnt 0x0" ::: "memory");
    __syncthreads();

    // ---- Compute: each thread does 4 pixels (column lx, rows lz+8*sub) ----
    const int lx = tid & 31;   // lanes of a wave cover consecutive x -> no LDS conflicts
    const int lz = tid >> 5;   // 0..7

    // Unnormalized separable Gaussian, sigma=1.5: exp(-d^2/4.5), d=-3..3.
    // The normalization constant cancels in the weighted-mean ratio.
    const float sw1d[7] = {0.13533528f, 0.41111229f, 0.80073740f, 1.0f,
                           0.80073740f, 0.41111229f, 0.13533528f};
    // exp(-50*x) == exp2(-50*log2(e)*x) -> single native v_exp_f32
    const float KC = -72.134752f;

    for (int sub = 0; sub < 4; ++sub) {
        const int ly = lz + sub * 8;
        const int cx = lx + PAD;
        const int cy = ly + PAD;
        const float c0 = smem[0 * LDSPLANE + cy * LDSD + cx];
        const float c1 = smem[1 * LDSPLANE + cy * LDSD + cx];
        const float c2 = smem[2 * LDSPLANE + cy * LDSD + cx];

        float acc0 = 0.0f, acc1 = 0.0f, acc2 = 0.0f, wsum = 0.0f;

#pragma unroll
        for (int ky = 0; ky < 7; ++ky) {
            const int   sy = ly + ky;
            const float wy = sw1d[ky];
#pragma unroll
            for (int kx = 0; kx < 7; ++kx) {
                const int sx = lx + kx;
                const float p0 = smem[0 * LDSPLANE + sy * LDSD + sx];
                const float p1 = smem[1 * LDSPLANE + sy * LDSD + sx];
                const float p2 = smem[2 * LDSPLANE + sy * LDSD + sx];
                const float l1 = fabsf(p0 - c0) + fabsf(p1 - c1) + fabsf(p2 - c2);
                // sw1d[ky]*sw1d[kx] folds to a literal after full unroll
                const float w = (wy * sw1d[kx]) * exp2f(KC * l1 * l1);
                wsum += w;
                acc0 = fmaf(w, p0, acc0);
                acc1 = fmaf(w, p1, acc1);
                acc2 = fmaf(w, p2, acc2);
            }
        }

        const float rw = __builtin_amdgcn_rcpf(wsum);
        const int ox = tileX0 + lx;
        const int oy = tileY0 + ly;
        size_t o = ((size_t)(b * CH + 0) * IMG_H + oy) * IMG_W + ox;
        out[o]                             = acc0 * rw;
        out[o + (size_t)IMG_H * IMG_W]     = acc1 * rw;
        out[o + (size_t)2 * IMG_H * IMG_W] = acc2 * rw;
    }
}

extern "C" void kernel_launch(void* const* d_in, const int* in_sizes, int n_in,
                              void* d_out, int out_size, void* d_ws, size_t ws_size,
                              hipStream_t stream) {
    const float* in  = (const float*)d_in[0];
    float*       out = (float*)d_out;
    const int B = in_sizes[0] / (CH * IMG_H * IMG_W);   // 8
    dim3 grid(IMG_W / TILE, IMG_H / TILE, B);           // 16 x 16 x 8 = 2048 blocks
    bilateral7x7_kernel<<<grid, 256, 0, stream>>>(in, out);
}